// MethodGCN_39616778338353
// MI455X (gfx1250) — compile-verified
//
#include <hip/hip_runtime.h>
#include <hip/hip_bf16.h>
#include <math.h>

// GCN 2-layer forward for MI455X (gfx1250, wave32, WMMA).
// Dense transforms use V_WMMA_F32_16X16X4_F32 (exact f32, matches reference).

#define N_NODES 100000
#define N_FEAT  512
#define N_HID   256
#define N_CLASS 10
#define N_EDGES 3200000

typedef __attribute__((ext_vector_type(2))) float v2f;
typedef __attribute__((ext_vector_type(8))) float v8f;

// ---------------------------------------------------------------------------
// f32 WMMA GEMM: C[M, NTILES*16 slab] = A[M,K] @ W[K,NOUT].
// Block = 256 threads = 8 waves covering 128 rows; each wave computes a
// 16 x (NTILES*16) strip, sharing one A fragment across NTILES WMMAs per
// k-step (cuts A HBM traffic NTILES-x). W is staged in LDS in K-chunks of KC
// (KC*16*NTILES floats), K-pair interleaved so each B fragment is one
// ds_load_b64.
// ---------------------------------------------------------------------------
template <int K, int KC, int NTILES, int NOUT, int OUT_STRIDE, bool PAD_N>
__global__ __launch_bounds__(256)
void gemm_wmma_f32(const float* __restrict__ A, const float* __restrict__ W,
                   float* __restrict__ C) {
  __shared__ float ldsW[KC * 16 * NTILES];

  const int nbase = blockIdx.y * (16 * NTILES);
  const int mbase = blockIdx.x * 128;
  const int tid   = threadIdx.x;

  const int wave  = tid >> 5;
  const int lane  = tid & 31;
  const int ln    = lane & 15;
  const int khalf = (lane >> 4) << 1;            // 0 for lanes 0-15, 2 for 16-31

  // A fragment row (clamped so EXEC stays all-1s through WMMA; store is guarded)
  const int row   = mbase + wave * 16 + ln;
  const int rowc  = row < N_NODES ? row : (N_NODES - 1);
  const float* __restrict__ arow = A + (size_t)rowc * K + khalf;

  v8f acc[NTILES] = {};

  for (int kc = 0; kc < K; kc += KC) {
    // Stage W[kc:kc+KC, nbase:nbase+16*NTILES] into LDS (K-pair interleaved).
    for (int i = tid; i < KC * 16 * NTILES; i += 256) {
      const int t = i / (KC * 16);
      const int r = i - t * (KC * 16);
      const int k = r >> 4, n = r & 15;
      const int col = nbase + t * 16 + n;
      float v;
      if (PAD_N) {
        v = (col < NOUT) ? W[(size_t)(kc + k) * NOUT + col] : 0.0f;
      } else {
        v = W[(size_t)(kc + k) * NOUT + col];
      }
      ldsW[t * (KC * 16) + (((k >> 1) * 16 + n) << 1) + (k & 1)] = v;
    }
    __syncthreads();

    const float* __restrict__ ap = arow + kc;
#pragma unroll 2
    for (int k = 0; k < KC; k += 4) {
      // A 16x4 fragment: lane<16 -> A[row][k..k+1], lane>=16 -> A[row][k+2..k+3]
      const v2f a = *(const v2f*)(ap + k);
      const int bidx = (((k + khalf) >> 1) * 16 + ln) << 1;
#pragma unroll
      for (int t = 0; t < NTILES; ++t) {
        const v2f b = *(const v2f*)&ldsW[t * (KC * 16) + bidx];
        acc[t] = __builtin_amdgcn_wmma_f32_16x16x4_f32(
            /*neg_a=*/false, a, /*neg_b=*/false, b,
            /*c_mod=*/(short)0, acc[t], /*reuse_a=*/false, /*reuse_b=*/false);
      }
    }
    __syncthreads();
  }

  // D layout: VGPR v -> M = v (lanes 0-15) or v+8 (lanes 16-31), N = ln
  const int srow0 = mbase + wave * 16 + ((lane >> 4) << 3);
  if (mbase + 128 <= N_NODES) {
    // Full block: unguarded stores (781 of 782 M-blocks take this path).
#pragma unroll
    for (int t = 0; t < NTILES; ++t) {
      float* __restrict__ p = C + (size_t)srow0 * OUT_STRIDE + nbase + t * 16 + ln;
#pragma unroll
      for (int v = 0; v < 8; ++v) p[(size_t)v * OUT_STRIDE] = acc[t][v];
    }
  } else {
#pragma unroll
    for (int t = 0; t < NTILES; ++t) {
#pragma unroll
      for (int v = 0; v < 8; ++v) {
        const int r = srow0 + v;
        if (r < N_NODES)
          C[(size_t)r * OUT_STRIDE + nbase + t * 16 + ln] = acc[t][v];
      }
    }
  }
}

// ---------------------------------------------------------------------------
// Edge-parallel SpMM (wide rows, F multiple of 128): agg[dst] += w*sup[src].
// One wave per edge; each lane handles a float4 (16B-aligned gather, hits L2
// since support1/h are L2-resident); per-f32 non-returning global atomics.
// ---------------------------------------------------------------------------
__global__ __launch_bounds__(256)
void spmm_edges_v4(const float* __restrict__ sup, float* __restrict__ agg,
                   const int* __restrict__ esrc, const int* __restrict__ edst,
                   const float* __restrict__ ew, int nfeat, int nedges) {
  const int e = blockIdx.x * 8 + (threadIdx.x >> 5);
  if (e >= nedges) return;
  const int lane = threadIdx.x & 31;
  const int s = esrc[e];
  const int d = edst[e];
  const float w = ew[e];
  const float4* __restrict__ srow = (const float4*)(sup + (size_t)s * nfeat);
  float* __restrict__ drow        = agg + (size_t)d * nfeat;
  for (int q = lane; q < (nfeat >> 2); q += 32) {
    const float4 v = srow[q];
    atomicAdd(&drow[q * 4 + 0], w * v.x);
    atomicAdd(&drow[q * 4 + 1], w * v.y);
    atomicAdd(&drow[q * 4 + 2], w * v.z);
    atomicAdd(&drow[q * 4 + 3], w * v.w);
  }
}

// Narrow-row SpMM for layer 2 (10 features, padded source stride 16).
__global__ __launch_bounds__(256)
void spmm_edges_narrow(const float* __restrict__ sup, int sup_stride,
                       float* __restrict__ agg, int agg_stride,
                       const int* __restrict__ esrc, const int* __restrict__ edst,
                       const float* __restrict__ ew, int nfeat, int nedges) {
  const int e = blockIdx.x * 8 + (threadIdx.x >> 5);
  if (e >= nedges) return;
  const int lane = threadIdx.x & 31;
  const int s = esrc[e];
  const int d = edst[e];
  const float w = ew[e];
  if (lane < nfeat) {
    atomicAdd(&agg[(size_t)d * agg_stride + lane],
              w * sup[(size_t)s * sup_stride + lane]);
  }
}

__global__ void zero_f32(float* __restrict__ p, size_t n) {
  const size_t i = (size_t)blockIdx.x * 256 + threadIdx.x;
  if (i < n) p[i] = 0.0f;
}

__global__ void bias_relu_256_v4(float* __restrict__ h, const float* __restrict__ b) {
  const size_t i = (size_t)blockIdx.x * 256 + threadIdx.x;   // float4 index
  if (i < (size_t)N_NODES * (N_HID / 4)) {
    float4* p = (float4*)h + i;
    float4 v = *p;
    const int f0 = (int)((i * 4) & (N_HID - 1));
    v.x = fmaxf(v.x + b[f0 + 0], 0.0f);
    v.y = fmaxf(v.y + b[f0 + 1], 0.0f);
    v.z = fmaxf(v.z + b[f0 + 2], 0.0f);
    v.w = fmaxf(v.w + b[f0 + 3], 0.0f);
    *p = v;
  }
}

__global__ void bias_logsoftmax10(float* __restrict__ out, const float* __restrict__ b2) {
  const int i = blockIdx.x * 256 + threadIdx.x;
  if (i >= N_NODES) return;
  float v[N_CLASS];
  float mx = -INFINITY;
#pragma unroll
  for (int c = 0; c < N_CLASS; ++c) {
    v[c] = out[(size_t)i * N_CLASS + c] + b2[c];
    mx = fmaxf(mx, v[c]);
  }
  float s = 0.0f;
#pragma unroll
  for (int c = 0; c < N_CLASS; ++c) s += __expf(v[c] - mx);
  const float lse = mx + __logf(s);
#pragma unroll
  for (int c = 0; c < N_CLASS; ++c) out[(size_t)i * N_CLASS + c] = v[c] - lse;
}

extern "C" void kernel_launch(void* const* d_in, const int* in_sizes, int n_in,
                              void* d_out, int out_size, void* d_ws, size_t ws_size,
                              hipStream_t stream) {
  const float* x    = (const float*)d_in[0];
  const int*   esrc = (const int*)  d_in[1];
  const int*   edst = (const int*)  d_in[2];
  const float* ew   = (const float*)d_in[3];
  const float* W1   = (const float*)d_in[4];
  const float* b1   = (const float*)d_in[5];
  const float* W2   = (const float*)d_in[6];
  const float* b2   = (const float*)d_in[7];
  float* out = (float*)d_out;

  // Workspace: support1 (102.4MB) | h (102.4MB). support2 reuses support1 region.
  char* ws = (char*)d_ws;
  float* support1 = (float*)ws;
  float* h        = (float*)(ws + (size_t)N_NODES * N_HID * sizeof(float));
  float* support2 = support1;  // [N_NODES, 16] padded (cols 10..15 zero)

  const int mtiles  = (N_NODES + 127) / 128;                 // 782
  const int eblocks = (N_EDGES + 7) / 8;                     // 400000
  const size_t hsz  = (size_t)N_NODES * N_HID;               // 25.6M
  const size_t osz  = (size_t)N_NODES * N_CLASS;             // 1.0M

  // ---- Layer 1: 64 cols per wave (NTILES=4), W staged in 128-K chunks ----
  gemm_wmma_f32<N_FEAT, 128, 4, N_HID, N_HID, false>
      <<<dim3(mtiles, N_HID / 64), 256, 0, stream>>>(x, W1, support1);
  zero_f32<<<(unsigned)((hsz + 255) / 256), 256, 0, stream>>>(h, hsz);
  spmm_edges_v4<<<eblocks, 256, 0, stream>>>(support1, h, esrc, edst, ew,
                                             N_HID, N_EDGES);
  bias_relu_256_v4<<<(unsigned)((hsz / 4 + 255) / 256), 256, 0, stream>>>(h, b1);

  // ---- Layer 2: N padded 10 -> 16, single chunk (K=256 in LDS at once) ----
  gemm_wmma_f32<N_HID, N_HID, 1, N_CLASS, 16, true>
      <<<dim3(mtiles, 1), 256, 0, stream>>>(h, W2, support2);
  zero_f32<<<(unsigned)((osz + 255) / 256), 256, 0, stream>>>(out, osz);
  spmm_edges_narrow<<<eblocks, 256, 0, stream>>>(support2, 16, out, N_CLASS,
                                                 esrc, edst, ew, N_CLASS, N_EDGES);
  bias_logsoftmax10<<<(N_NODES + 255) / 256, 256, 0, stream>>>(out, b2);
}